// ProbHypernet_27504970564169
// MI455X (gfx1250) — compile-verified
//
#include <hip/hip_runtime.h>
#include <hip/hip_bf16.h>

typedef __attribute__((ext_vector_type(2))) float v2f;
typedef __attribute__((ext_vector_type(4))) float v4f;
typedef __attribute__((ext_vector_type(8))) float v8f;

#define MB    64      // batch
#define IND   1023    // IN_DIM
#define RD    1024    // R = IN_DIM+1
#define HD    256     // H_DIM
#define HR    257     // H_DIM+1
#define OD    1024    // OUT_DIM

// ---------------------------------------------------------------------------
// zero accumulation buffers (hacc [64*256]) and the KL slot of d_out
// ---------------------------------------------------------------------------
__global__ void zero_kernel(float* __restrict__ hacc, float* __restrict__ out_kl) {
    int i = blockIdx.x * 256 + threadIdx.x;
    if (i < MB * HD) hacc[i] = 0.0f;
    if (i == 0) *out_kl = 0.0f;
}

// ---------------------------------------------------------------------------
// d_kl: pure parameter reduction, one block per VMG layer
// d = 0.5*(sum(e^li)*sum(e^lo) + sum(mu^2) - r*c - c*sum(li) - r*sum(lo))
// ---------------------------------------------------------------------------
__global__ void kl_kernel(const float* __restrict__ muA, const float* __restrict__ liA, const float* __restrict__ loA,
                          const float* __restrict__ muB, const float* __restrict__ liB, const float* __restrict__ loB,
                          const float* __restrict__ muC, const float* __restrict__ liC, const float* __restrict__ loC,
                          const float* __restrict__ muD, const float* __restrict__ liD, const float* __restrict__ loD,
                          float* __restrict__ out_kl) {
    const float* mu; const float* li; const float* lo; int r, c;
    switch (blockIdx.x) {
        case 0:  mu = muA; li = liA; lo = loA; r = RD; c = HD; break;
        case 1:  mu = muB; li = liB; lo = loB; r = HR; c = RD; break;
        case 2:  mu = muC; li = liC; lo = loC; r = HR; c = RD; break;
        default: mu = muD; li = liD; lo = loD; r = HR; c = OD; break;
    }
    const int t = threadIdx.x;
    float s_mu2 = 0.f, s_ie = 0.f, s_i = 0.f, s_oe = 0.f, s_o = 0.f;
    const int n_mu = r * c;
    for (int idx = t; idx < n_mu; idx += 256) { float m0 = mu[idx]; s_mu2 += m0 * m0; }
    for (int idx = t; idx < r; idx += 256)    { float lv = li[idx]; s_ie += __expf(lv); s_i += lv; }
    for (int idx = t; idx < c; idx += 256)    { float lv = lo[idx]; s_oe += __expf(lv); s_o += lv; }
    __shared__ float sA[256], sB[256], sC[256], sD[256], sE[256];
    sA[t] = s_mu2; sB[t] = s_ie; sC[t] = s_i; sD[t] = s_oe; sE[t] = s_o;
    __syncthreads();
    for (int off = 128; off > 0; off >>= 1) {
        if (t < off) {
            sA[t] += sA[t + off]; sB[t] += sB[t + off]; sC[t] += sC[t + off];
            sD[t] += sD[t + off]; sE[t] += sE[t + off];
        }
        __syncthreads();
    }
    if (t == 0) {
        float d = 0.5f * (sB[0] * sD[0] + sA[0] - (float)r * (float)c
                          - (float)c * sC[0] - (float)r * sE[0]);
        atomicAdd(out_kl, d);
    }
}

// ---------------------------------------------------------------------------
// Layer 1 (xh): hacc[m,c] += sum_{r in chunk} ( xa*mu_xh[r,c] + sc[c]*xa*sr[r]*E_xh[m,r,c] )
// grid (64, 8), block 256 threads (thread = c), r-chunk = 128.
// E_xh is single-use: non-temporal loads keep it out of L2's working set.
// ---------------------------------------------------------------------------
__global__ void hyper_xh_kernel(const float* __restrict__ x,
                                const float* __restrict__ mu_xh,
                                const float* __restrict__ li_xh,
                                const float* __restrict__ lo_xh,
                                const float* __restrict__ E_xh,
                                float* __restrict__ hacc) {
    __shared__ float aS[128], bS[128];
    const int m  = blockIdx.x;
    const int r0 = blockIdx.y * 128;
    const int c  = threadIdx.x;
    if (threadIdx.x < 128) {
        int r = r0 + threadIdx.x;
        float xv = (r < IND) ? x[m * IND + r] : 1.0f;   // bias augment
        bS[threadIdx.x] = xv;
        aS[threadIdx.x] = xv * __expf(0.5f * li_xh[r]);
    }
    __syncthreads();
    const float* __restrict__ Mp = mu_xh + (size_t)r0 * HD + c;
    const float* __restrict__ Ep = E_xh + ((size_t)m * RD + r0) * HD + c;
    float accB = 0.f, accE = 0.f;
    #pragma unroll 4
    for (int i = 0; i < 128; ++i) {
        accB += bS[i] * Mp[(size_t)i * HD];                               // reused: RT
        accE += aS[i] * __builtin_nontemporal_load(Ep + (size_t)i * HD);  // stream: NT
    }
    float sc = __expf(0.5f * lo_xh[c]);
    atomicAdd(&hacc[m * HD + c], accB + sc * accE);
}

__global__ void tanh_kernel(const float* __restrict__ hacc, float* __restrict__ h) {
    int i = blockIdx.x * 256 + threadIdx.x;   // 64*256 total
    h[i] = tanhf(hacc[i]);
}

// ---------------------------------------------------------------------------
// 3 hyper heads: out[m,j] = sum_k ha*mu[k,j] + sc[j]*sum_k ha*srk[k]*E[m,k,j]
// Directly produce what the main layer needs:
//   head0 (hmu): u[m,j]  = xa(m,j) * out
//   head1 (hli): v[m,j]  = xa(m,j) * exp(0.5*out)
//   head2 (hlo): scm[m,j]= exp(0.5*out)
// grid (64, 3), block 256 threads, 4 cols (v4f) per thread; E loads NT.
// ---------------------------------------------------------------------------
__global__ void hyper_heads_kernel(const float* __restrict__ h, const float* __restrict__ x,
                                   const float* __restrict__ mu0, const float* __restrict__ li0,
                                   const float* __restrict__ lo0, const float* __restrict__ E0,
                                   const float* __restrict__ mu1, const float* __restrict__ li1,
                                   const float* __restrict__ lo1, const float* __restrict__ E1,
                                   const float* __restrict__ mu2, const float* __restrict__ li2,
                                   const float* __restrict__ lo2, const float* __restrict__ E2,
                                   float* __restrict__ u, float* __restrict__ vv,
                                   float* __restrict__ scm) {
    __shared__ float haS[HR], harS[HR];
    const int m = blockIdx.x;
    const int head = blockIdx.y;
    const float* mu; const float* li; const float* lo; const float* E;
    if (head == 0)      { mu = mu0; li = li0; lo = lo0; E = E0; }
    else if (head == 1) { mu = mu1; li = li1; lo = lo1; E = E1; }
    else                { mu = mu2; li = li2; lo = lo2; E = E2; }
    const int t = threadIdx.x;
    for (int k = t; k < HR; k += 256) {
        float hv = (k < HD) ? h[m * HD + k] : 1.0f;    // bias augment
        haS[k]  = hv;
        harS[k] = hv * __expf(0.5f * li[k]);
    }
    __syncthreads();
    const v4f* __restrict__ Mp = (const v4f*)mu + t;                       // [HR, 1024/4]
    const v4f* __restrict__ Ep = (const v4f*)E + (size_t)m * HR * 256 + t; // [64, HR, 1024/4]
    v4f aM = {0.f, 0.f, 0.f, 0.f}, aE = {0.f, 0.f, 0.f, 0.f};
    for (int k = 0; k < HR; ++k) {
        v4f mv = Mp[(size_t)k * 256];                                   // reused: RT
        v4f ev = __builtin_nontemporal_load(Ep + (size_t)k * 256);      // stream: NT
        float b = haS[k], a = harS[k];
        aM += b * mv;
        aE += a * ev;
    }
    const int j = t * 4;
    float o0 = aM.x + __expf(0.5f * lo[j + 0]) * aE.x;
    float o1 = aM.y + __expf(0.5f * lo[j + 1]) * aE.y;
    float o2 = aM.z + __expf(0.5f * lo[j + 2]) * aE.z;
    float o3 = aM.w + __expf(0.5f * lo[j + 3]) * aE.w;
    float xa0 = (j + 0 < IND) ? x[m * IND + j + 0] : 1.0f;
    float xa1 = (j + 1 < IND) ? x[m * IND + j + 1] : 1.0f;
    float xa2 = (j + 2 < IND) ? x[m * IND + j + 2] : 1.0f;
    float xa3 = (j + 3 < IND) ? x[m * IND + j + 3] : 1.0f;
    if (head == 0) {
        u[m * RD + j + 0] = xa0 * o0; u[m * RD + j + 1] = xa1 * o1;
        u[m * RD + j + 2] = xa2 * o2; u[m * RD + j + 3] = xa3 * o3;
    } else if (head == 1) {
        vv[m * RD + j + 0] = xa0 * __expf(0.5f * o0);
        vv[m * RD + j + 1] = xa1 * __expf(0.5f * o1);
        vv[m * RD + j + 2] = xa2 * __expf(0.5f * o2);
        vv[m * RD + j + 3] = xa3 * __expf(0.5f * o3);
    } else {
        scm[m * OD + j + 0] = __expf(0.5f * o0);
        scm[m * OD + j + 1] = __expf(0.5f * o1);
        scm[m * OD + j + 2] = __expf(0.5f * o2);
        scm[m * OD + j + 3] = __expf(0.5f * o3);
    }
}

// ---------------------------------------------------------------------------
// Main-layer mean term via f32 WMMA: out[64,1024] = u[64,1024] @ M_main[1024,1024]
// One 16x16 output tile per wave; v_wmma_f32_16x16x4_f32, K-loop of 256 steps.
// A(16x4): lane = M (lane&15), VGPR j -> K = j + 2*(lane>=16)
// B(4x16): lane = N (lane&15), VGPR j -> K = j + 2*(lane>=16)
// C/D:     VGPR i -> M = i + 8*(lane>=16), N = lane&15
// grid (4 m-tiles, 8), 256 threads = 8 waves (one n-tile each); EXEC all ones.
// u + M_main stay L2-resident (E streams are marked NT elsewhere).
// ---------------------------------------------------------------------------
__global__ void main_gemm_wmma_kernel(const float* __restrict__ u,
                                      const float* __restrict__ Mmain,
                                      float* __restrict__ out) {
    const int lane  = threadIdx.x & 31;
    const int wave  = threadIdx.x >> 5;
    const int mbase = blockIdx.x * 16;
    const int cbase = (blockIdx.y * 8 + wave) * 16;
    const int l15 = lane & 15;
    const int hi  = lane >> 4;            // 0 or 1
    const int ma  = mbase + l15;          // A row held by this lane
    const int nb  = cbase + l15;          // B col held by this lane
    v8f acc = {};
    for (int r0 = 0; r0 < RD; r0 += 4) {
        const int k0 = r0 + 2 * hi;
        v2f a, b;
        a.x = u[(size_t)ma * RD + k0];
        a.y = u[(size_t)ma * RD + k0 + 1];
        b.x = Mmain[(size_t)(k0)     * OD + nb];
        b.y = Mmain[(size_t)(k0 + 1) * OD + nb];
        acc = __builtin_amdgcn_wmma_f32_16x16x4_f32(
            /*neg_a=*/false, a, /*neg_b=*/false, b,
            /*c_mod=*/(short)0, acc, /*reuse_a=*/false, /*reuse_b=*/false);
    }
    #pragma unroll
    for (int i = 0; i < 8; ++i) {
        int mrow = mbase + i + 8 * hi;
        out[(size_t)mrow * OD + cbase + l15] = acc[i];
    }
}

// ---------------------------------------------------------------------------
// Main-layer noise term: out[m,c] += scm[m,c] * sum_{r in chunk} v[m,r]*E_main[m,r,c]
// grid (64, 8), 256 threads, r-chunk = 128, v4f along c (fully coalesced).
// Dominant kernel: streams 256 MB of E_main at HBM rate; NT loads + prefetch.
// ---------------------------------------------------------------------------
__global__ void main_evm_kernel(const float* __restrict__ v,
                                const float* __restrict__ scm,
                                const float* __restrict__ E_main,
                                float* __restrict__ out) {
    __shared__ float vS[128];
    const int m  = blockIdx.x;
    const int r0 = blockIdx.y * 128;
    const int t  = threadIdx.x;
    if (t < 128) vS[t] = v[m * RD + r0 + t];
    __syncthreads();
    const v4f* __restrict__ Ep = (const v4f*)E_main + ((size_t)m * RD + r0) * 256 + t;
    v4f acc = {0.f, 0.f, 0.f, 0.f};
    #pragma unroll 4
    for (int i = 0; i < 128; ++i) {
        if ((i & 31) == 0) {                // uniform branch; global_prefetch_b8 ahead
            int ip = i + 32; if (ip > 127) ip = 127;
            __builtin_prefetch(Ep + (size_t)ip * 256, 0, 0);
        }
        v4f ev = __builtin_nontemporal_load(Ep + (size_t)i * 256);   // stream: NT
        acc += vS[i] * ev;
    }
    const int j = t * 4;
    const float* sp = scm + (size_t)m * OD + j;
    atomicAdd(&out[(size_t)m * OD + j + 0], sp[0] * acc.x);
    atomicAdd(&out[(size_t)m * OD + j + 1], sp[1] * acc.y);
    atomicAdd(&out[(size_t)m * OD + j + 2], sp[2] * acc.z);
    atomicAdd(&out[(size_t)m * OD + j + 3], sp[3] * acc.w);
}

// ---------------------------------------------------------------------------
extern "C" void kernel_launch(void* const* d_in, const int* in_sizes, int n_in,
                              void* d_out, int out_size, void* d_ws, size_t ws_size,
                              hipStream_t stream) {
    const float* x      = (const float*)d_in[0];
    const float* M_main = (const float*)d_in[1];
    const float* mu_xh  = (const float*)d_in[2];
    const float* li_xh  = (const float*)d_in[3];
    const float* lo_xh  = (const float*)d_in[4];
    const float* mu_hmu = (const float*)d_in[5];
    const float* li_hmu = (const float*)d_in[6];
    const float* lo_hmu = (const float*)d_in[7];
    const float* mu_hli = (const float*)d_in[8];
    const float* li_hli = (const float*)d_in[9];
    const float* lo_hli = (const float*)d_in[10];
    const float* mu_hlo = (const float*)d_in[11];
    const float* li_hlo = (const float*)d_in[12];
    const float* lo_hlo = (const float*)d_in[13];
    const float* E_xh   = (const float*)d_in[14];
    const float* E_hmu  = (const float*)d_in[15];
    const float* E_hli  = (const float*)d_in[16];
    const float* E_hlo  = (const float*)d_in[17];
    const float* E_main = (const float*)d_in[18];

    float* out    = (float*)d_out;          // [64,1024] then KL scalar at [65536]
    float* out_kl = out + MB * OD;

    float* W    = (float*)d_ws;
    float* hacc = W;                        // 64*256
    float* h    = W + MB * HD;              // 64*256
    float* u    = W + 2 * MB * HD;          // 64*1024
    float* vv   = u + MB * RD;              // 64*1024
    float* scm  = vv + MB * RD;             // 64*1024

    zero_kernel<<<dim3((MB * HD + 255) / 256), dim3(256), 0, stream>>>(hacc, out_kl);

    kl_kernel<<<dim3(4), dim3(256), 0, stream>>>(
        mu_xh, li_xh, lo_xh, mu_hmu, li_hmu, lo_hmu,
        mu_hli, li_hli, lo_hli, mu_hlo, li_hlo, lo_hlo, out_kl);

    hyper_xh_kernel<<<dim3(MB, 8), dim3(256), 0, stream>>>(
        x, mu_xh, li_xh, lo_xh, E_xh, hacc);

    tanh_kernel<<<dim3(MB), dim3(256), 0, stream>>>(hacc, h);

    hyper_heads_kernel<<<dim3(MB, 3), dim3(256), 0, stream>>>(
        h, x,
        mu_hmu, li_hmu, lo_hmu, E_hmu,
        mu_hli, li_hli, lo_hli, E_hli,
        mu_hlo, li_hlo, lo_hlo, E_hlo,
        u, vv, scm);

    main_gemm_wmma_kernel<<<dim3(4, 8), dim3(256), 0, stream>>>(u, M_main, out);

    main_evm_kernel<<<dim3(MB, 8), dim3(256), 0, stream>>>(vv, scm, E_main, out);
}